// Top1Gate_15796889714905
// MI455X (gfx1250) — compile-verified
//
#include <hip/hip_runtime.h>

typedef float v2f __attribute__((ext_vector_type(2)));
typedef float v4f __attribute__((ext_vector_type(4)));
typedef float v8f __attribute__((ext_vector_type(8)));

constexpr int S   = 8192;   // tokens
constexpr int D   = 2048;   // model dim
constexpr int E   = 64;     // experts
constexpr int CAP = 128;    // capacity = ceil(S/E) * 1.0
constexpr int TPB = 128;    // tokens per block in k_gate
constexpr int NBLK = S / TPB;  // 64

// ---------------------------------------------------------------------------
// Kernel 1: logits GEMM (fp32 WMMA 16x16x4) + softmax + top1 + block partials
// ---------------------------------------------------------------------------
__global__ __launch_bounds__(256) void k_gate(const float* __restrict__ x,
                                              const float* __restrict__ wg,
                                              int*   __restrict__ idxArr,
                                              int*   __restrict__ rankArr,
                                              float* __restrict__ gateArr,
                                              int*   __restrict__ cnt,
                                              float* __restrict__ mePart)
{
    __shared__ float lg[TPB * E];       // 32 KB: logits, later normalized gates
    __shared__ int   sIdx[TPB];

    const int tid  = threadIdx.x;
    const int wave = tid >> 5;
    const int lane = tid & 31;
    const int blk  = blockIdx.x;

    // A (16x4 f32) layout: lanes 0-15 rows 0-15 hold K={0,1}; lanes 16-31 hold K={2,3}
    // B (4x16 f32) layout: lanes 0-15 cols hold K={0,1}; lanes 16-31 K={2,3}
    const int m  = lane & 15;
    const int kk = (lane >> 4) << 1;
    const int tokBase = blk * TPB + wave * 16;

    const float* aRow  = x  + (size_t)(tokBase + m) * D + kk;
    const float* bRow0 = wg + (size_t)( 0 + m) * D + kk;
    const float* bRow1 = wg + (size_t)(16 + m) * D + kk;
    const float* bRow2 = wg + (size_t)(32 + m) * D + kk;
    const float* bRow3 = wg + (size_t)(48 + m) * D + kk;

    v8f acc0 = {}, acc1 = {}, acc2 = {}, acc3 = {};
    for (int k0 = 0; k0 < D; k0 += 4) {
        v2f a  = *(const v2f*)(aRow  + k0);
        v2f b0 = *(const v2f*)(bRow0 + k0);
        v2f b1 = *(const v2f*)(bRow1 + k0);
        v2f b2 = *(const v2f*)(bRow2 + k0);
        v2f b3 = *(const v2f*)(bRow3 + k0);
        acc0 = __builtin_amdgcn_wmma_f32_16x16x4_f32(false, a, false, b0, (short)0, acc0, false, false);
        acc1 = __builtin_amdgcn_wmma_f32_16x16x4_f32(false, a, false, b1, (short)0, acc1, false, false);
        acc2 = __builtin_amdgcn_wmma_f32_16x16x4_f32(false, a, false, b2, (short)0, acc2, false, false);
        acc3 = __builtin_amdgcn_wmma_f32_16x16x4_f32(false, a, false, b3, (short)0, acc3, false, false);
    }

    // C/D layout: VGPR j, lanes 0-15 -> M=j, N=lane ; lanes 16-31 -> M=j+8, N=lane-16
    const int n   = lane & 15;
    const int mhi = (lane >> 4) << 3;
    #pragma unroll
    for (int j = 0; j < 8; ++j) {
        const int tok = wave * 16 + mhi + j;
        lg[tok * E +  0 + n] = acc0[j];
        lg[tok * E + 16 + n] = acc1[j];
        lg[tok * E + 32 + n] = acc2[j];
        lg[tok * E + 48 + n] = acc3[j];
    }
    __syncthreads();

    // Per-token softmax + top1 (first-max-wins, matching jnp.argmax)
    if (tid < TPB) {
        const int t = tid;
        float mx = -3.402823466e38f; int am = 0;
        for (int e = 0; e < E; ++e) { float v = lg[t*E+e]; if (v > mx) { mx = v; am = e; } }
        float sum = 0.f;
        for (int e = 0; e < E; ++e) { float g = __expf(lg[t*E+e] - mx); sum += g; lg[t*E+e] = g; }
        const float inv = 1.f / sum;
        for (int e = 0; e < E; ++e) lg[t*E+e] *= inv;   // normalized gates in LDS
        const int s = blk * TPB + t;
        gateArr[s] = inv;          // gate of argmax expert = exp(0)/sum
        idxArr[s]  = am;
        sIdx[t]    = am;
    }
    __syncthreads();

    // Within-block rank in token order (deterministic serial-scan semantics)
    if (tid < TPB) {
        const int t  = tid;
        const int am = sIdx[t];
        int r = 0;
        for (int t2 = 0; t2 < t; ++t2) r += (sIdx[t2] == am) ? 1 : 0;
        rankArr[blk * TPB + t] = r;
    }
    // Per-block expert histogram + gate-sum partials (for l_aux)
    if (tid < E) {
        int c = 0; float ms = 0.f;
        for (int t = 0; t < TPB; ++t) {
            c  += (sIdx[t] == tid) ? 1 : 0;
            ms += lg[t * E + tid];
        }
        cnt[blk * E + tid]    = c;
        mePart[blk * E + tid] = ms;
    }
}

// ---------------------------------------------------------------------------
// Kernel 2: exclusive scan of block histograms per expert + l_aux
// ---------------------------------------------------------------------------
__global__ __launch_bounds__(64) void k_scan(const int*   __restrict__ cnt,
                                             const float* __restrict__ mePart,
                                             int*   __restrict__ off,
                                             float* __restrict__ out)
{
    __shared__ float red[E];
    const int e = threadIdx.x;      // one expert per lane, 2 waves
    int run = 0; float ms = 0.f;
    for (int b = 0; b < NBLK; ++b) {
        off[b * E + e] = run;
        run += cnt[b * E + e];
        ms  += mePart[b * E + e];
    }
    const float me = ms / (float)S;
    const float ce = (float)run / (float)S;   // mask mean BEFORE capacity drop
    red[e] = me * ce;
    __syncthreads();
    if (e == 0) {
        float s = 0.f;
        for (int i = 0; i < E; ++i) s += red[i];
        out[0] = s * (float)E;     // mean_e(me*ce)*E*E = E * sum_e(me*ce)
    }
}

// ---------------------------------------------------------------------------
// Kernel 3: write combine [S,E,C] + dispatch mask; NT zero-fill + 1 nonzero/row
// ---------------------------------------------------------------------------
__global__ __launch_bounds__(256) void k_fill(const int*   __restrict__ idxArr,
                                              const int*   __restrict__ rankArr,
                                              const float* __restrict__ gateArr,
                                              const int*   __restrict__ off,
                                              float* __restrict__ out)
{
    const int s = blockIdx.x;
    const int t = threadIdx.x;
    const int e   = idxArr[s];
    const int loc = off[(s >> 7) * E + e] + rankArr[s];
    const bool keep = (loc < CAP);
    const float g = gateArr[s];
    const int f  = e * CAP + loc;     // flat nonzero index within the row
    const int f4 = f >> 2, fi = f & 3;

    v4f* crow = (v4f*)(out + 1) + (size_t)s * (E * CAP / 4);
    v4f* mrow = crow + (size_t)S * (E * CAP / 4);
    #pragma unroll
    for (int i = 0; i < 8; ++i) {
        const int q = t + i * 256;            // float4 index within 2048-wide row
        v4f v  = {0.f, 0.f, 0.f, 0.f};
        v4f mv = {0.f, 0.f, 0.f, 0.f};
        if (keep && q == f4) { v[fi] = g; mv[fi] = 1.0f; }
        __builtin_nontemporal_store(v,  crow + q);    // 537 MB >> 192 MB L2 -> NT
        __builtin_nontemporal_store(mv, mrow + q);
    }
}

// ---------------------------------------------------------------------------
extern "C" void kernel_launch(void* const* d_in, const int* in_sizes, int n_in,
                              void* d_out, int out_size, void* d_ws, size_t ws_size,
                              hipStream_t stream)
{
    const float* x  = (const float*)d_in[0];   // [S, D]
    const float* wg = (const float*)d_in[1];   // [E, D]
    float* out = (float*)d_out;

    char* ws = (char*)d_ws;
    int*   idxArr  = (int*)ws;    ws += (size_t)S * sizeof(int);
    int*   rankArr = (int*)ws;    ws += (size_t)S * sizeof(int);
    float* gateArr = (float*)ws;  ws += (size_t)S * sizeof(float);
    int*   cnt     = (int*)ws;    ws += (size_t)NBLK * E * sizeof(int);
    float* mePart  = (float*)ws;  ws += (size_t)NBLK * E * sizeof(float);
    int*   off     = (int*)ws;    ws += (size_t)NBLK * E * sizeof(int);

    k_gate<<<NBLK, 256, 0, stream>>>(x, wg, idxArr, rankArr, gateArr, cnt, mePart);
    k_scan<<<1, 64, 0, stream>>>(cnt, mePart, off, out);
    k_fill<<<S, 256, 0, stream>>>(idxArr, rankArr, gateArr, off, out);
}